// Upsample_66640712565473
// MI455X (gfx1250) — compile-verified
//
#include <hip/hip_runtime.h>
#include <stdint.h>

// ---------------------------------------------------------------------------
// 2x upsample (upfirdn2d, up=2, pad0=2, pad1=1, 4x4 separable kernel)
// x: (N=8, C=64, H=256, W=256) f32  ->  out: (N, C, 2H, 2W) f32
//
// Memory-bound: ~0.67 GB traffic -> ~29 us floor at 23.3 TB/s.
// CDNA5 path used: async global->LDS tile staging (ASYNCcnt) + b128 NT stores.
// ---------------------------------------------------------------------------

#ifndef __has_builtin
#define __has_builtin(x) 0
#endif

#if __has_builtin(__builtin_amdgcn_global_load_async_to_lds_b32)
#define HAVE_ASYNC_LDS 1
#else
#define HAVE_ASYNC_LDS 0
#endif

typedef __attribute__((ext_vector_type(4))) float v4f;

#define TW 64              // input tile width per block
#define TH 16              // input tile height per block
#define LW (TW + 2)        // LDS row length incl. halo (66)
#define LH (TH + 2)        // LDS rows incl. halo (18)
#define NTHREADS 256

__device__ __forceinline__ void wait_async_lds() {
#if HAVE_ASYNC_LDS
#if __has_builtin(__builtin_amdgcn_s_wait_asynccnt)
  __builtin_amdgcn_s_wait_asynccnt(0);
#else
  asm volatile("s_wait_asynccnt 0" ::: "memory");
#endif
#endif
}

__global__ __launch_bounds__(NTHREADS)
void Upsample2x_kernel(const float* __restrict__ x,
                       const float* __restrict__ k2,   // 4x4 row-major
                       float* __restrict__ out,
                       int H, int W)
{
  __shared__ float smem[LH * LW];

  const int tid    = threadIdx.x;
  const int tileX0 = blockIdx.x * TW;
  const int tileY0 = blockIdx.y * TH;
  const int nc     = blockIdx.z;

  const float* __restrict__ img  = x   + (size_t)nc * H * W;
  float*       __restrict__ oimg = out + (size_t)nc * (2 * H) * (2 * W);

  // ---- Stage input tile + 1-halo into LDS ------------------------------
  // In-bounds: async global->LDS copy (no VGPR round-trip).
  // Out-of-bounds (zero padding): ds store of 0.  Disjoint LDS addresses,
  // so the two counters (ASYNCcnt / DScnt) never race on the same word.
  for (int e = tid; e < LH * LW; e += NTHREADS) {
    const int lr = e / LW;
    const int lc = e - lr * LW;
    const int gr = tileY0 + lr - 1;
    const int gc = tileX0 + lc - 1;
    const bool valid = ((unsigned)gr < (unsigned)H) && ((unsigned)gc < (unsigned)W);
    if (valid) {
#if HAVE_ASYNC_LDS
      __builtin_amdgcn_global_load_async_to_lds_b32(
          (int*)(img + (size_t)gr * W + gc), (int*)&smem[e], 0, 0);
#else
      smem[e] = img[(size_t)gr * W + gc];
#endif
    } else {
      smem[e] = 0.0f;
    }
  }
  wait_async_lds();       // drain this wave's ASYNCcnt before barrier
  __syncthreads();        // compiler adds s_wait_dscnt for the ds stores

  // ---- Filter weights (uniform address -> scalar loads) ----------------
  const float w00 = k2[0],  w01 = k2[1],  w02 = k2[2],  w03 = k2[3];
  const float w10 = k2[4],  w11 = k2[5],  w12 = k2[6],  w13 = k2[7];
  const float w20 = k2[8],  w21 = k2[9],  w22 = k2[10], w23 = k2[11];
  const float w30 = k2[12], w31 = k2[13], w32 = k2[14], w33 = k2[15];

  // Each thread: 2 input cols (c0, c0+1) x 2 input rows -> four float4 stores
  const int txi = tid & 31;           // 32 column-pairs across the 64-wide tile
  const int tyi = tid >> 5;           // 8 row slots
  const int c0  = 2 * txi;

#pragma unroll
  for (int rs = 0; rs < 2; ++rs) {
    const int r = tyi + rs * 8;       // input row within tile, 0..15

    // halo-shifted row pointers: sm1 = row r-1, s0 = row r, sp1 = row r+1,
    // each starting at tile column c0-1.
    const float* sm1 = &smem[(r + 0) * LW + c0];
    const float* s0  = &smem[(r + 1) * LW + c0];
    const float* sp1 = &smem[(r + 2) * LW + c0];

    const float a0 = sm1[0], a1 = sm1[1], a2 = sm1[2], a3 = sm1[3]; // row r-1
    const float b0 = s0[0],  b1 = s0[1],  b2 = s0[2],  b3 = s0[3];  // row r
    const float g0 = sp1[0], g1 = sp1[1], g2 = sp1[2], g3 = sp1[3]; // row r+1

    // even output row 2r: vertical taps (r-1 -> k2 row0, r -> k2 row2)
    v4f ev;
    ev.x = w00 * a0 + w02 * a1 + w20 * b0 + w22 * b1;   // col 2c0
    ev.y = w01 * a1 + w03 * a2 + w21 * b1 + w23 * b2;   // col 2c0+1
    ev.z = w00 * a1 + w02 * a2 + w20 * b1 + w22 * b2;   // col 2c0+2
    ev.w = w01 * a2 + w03 * a3 + w21 * b2 + w23 * b3;   // col 2c0+3

    // odd output row 2r+1: vertical taps (r -> k2 row1, r+1 -> k2 row3)
    v4f od;
    od.x = w10 * b0 + w12 * b1 + w30 * g0 + w32 * g1;
    od.y = w11 * b1 + w13 * b2 + w31 * g1 + w33 * g2;
    od.z = w10 * b1 + w12 * b2 + w30 * g1 + w32 * g2;
    od.w = w11 * b2 + w13 * b3 + w31 * g2 + w33 * g3;

    const int orow = (tileY0 + r) * 2;
    const int ocol = (tileX0 + c0) * 2;                  // multiple of 4 -> 16B aligned
    const size_t ow = (size_t)(2 * W);

    // Output is write-once and >> L2 (537 MB vs 192 MB): nontemporal b128 stores.
    __builtin_nontemporal_store(ev, (v4f*)(oimg + (size_t)orow * ow + ocol));
    __builtin_nontemporal_store(od, (v4f*)(oimg + (size_t)(orow + 1) * ow + ocol));
  }
}

extern "C" void kernel_launch(void* const* d_in, const int* in_sizes, int n_in,
                              void* d_out, int out_size, void* d_ws, size_t ws_size,
                              hipStream_t stream) {
  (void)n_in; (void)out_size; (void)d_ws; (void)ws_size;

  const float* x  = (const float*)d_in[0];   // (8,64,256,256) f32
  const float* k2 = (const float*)d_in[1];   // 4x4 f32
  float* out      = (float*)d_out;           // (8,64,512,512) f32

  const int H = 256, W = 256;
  const int NC = in_sizes[0] / (H * W);      // 8*64 = 512 images

  dim3 grid(W / TW, H / TH, NC);             // (4, 16, 512)
  Upsample2x_kernel<<<grid, NTHREADS, 0, stream>>>(x, k2, out, H, W);
}